// Block2_53231824666625
// MI455X (gfx1250) — compile-verified
//
#include <hip/hip_runtime.h>

typedef float v2f __attribute__((ext_vector_type(2)));
typedef float v8f __attribute__((ext_vector_type(8)));
typedef unsigned int u32x4 __attribute__((ext_vector_type(4)));
typedef int i32x4 __attribute__((ext_vector_type(4)));
typedef int i32x8 __attribute__((ext_vector_type(8)));

#define MN 12  // neighbors per atom

__device__ __forceinline__ float sig_(float x) { return 1.0f / (1.0f + expf(-x)); }
__device__ __forceinline__ float sp_(float x)  { return fmaxf(x, 0.0f) + log1pf(expf(-fabsf(x))); }

// ---------------------------------------------------------------------------
// TDM 1-D load: n_dwords f32 elements, global -> LDS, optional LDS padding of
// 1 DWORD after every 32 DWORDs (pad_interval=4 -> 32 DW, pad_amount=0 -> 1 DW)
// to produce a bank-conflict-free [.][33] layout.  D# built per CDNA5 ISA
// CDNA5 ISA (group0: count|lds|global|type=2, group1: dims/strides).
// This toolchain exposes the 6-arg builtin: (g0, g1, g2, g3, g4, cpol).
// ---------------------------------------------------------------------------
__device__ __forceinline__ void tdm_load_1d(unsigned int lds_addr, const void* gptr,
                                            unsigned int n_dwords, int pad_en) {
  unsigned long long ga = (unsigned long long)(uintptr_t)gptr;
  u32x4 g0 = {
    1u,                                                    // count=1 (valid user D#)
    lds_addr,                                              // lds_addr (bytes)
    (unsigned int)ga,                                      // global_addr[31:0]
    (((unsigned int)(ga >> 32)) & 0x1FFFFFFu) | 0x80000000u // global_addr[56:32] | type=2
  };
  int w0 = (2 << 16) | (pad_en ? ((1 << 20) | (4 << 22)) : 0); // data_size=4B, pad cfg
  i32x8 g1 = {
    w0,
    (int)((n_dwords & 0xFFFFu) << 16),       // tensor_dim0[15:0]  (bits 63:48)
    (int)((n_dwords >> 16) & 0xFFFFu),       // tensor_dim0[31:16]; tensor_dim1 lo = 0
    (int)((n_dwords & 0xFFFFu) << 16),       // tile_dim0 (bits 127:112); tensor_dim1 hi = 0
    0,                                       // tile_dim1=0, tile_dim2=0 (1-D)
    (int)n_dwords,                           // tensor_dim0_stride lo
    0, 0
  };
  i32x4 z4 = {0, 0, 0, 0};
  i32x8 z8 = {0, 0, 0, 0, 0, 0, 0, 0};
  __builtin_amdgcn_tensor_load_to_lds(g0, g1, z4, z4, z8, 0);
}

// ---------------------------------------------------------------------------
// Kernel A: gated message pass.  Block = 4 waves, handles 16 atoms.
// GEMM (192 rows x 112) @ (112 x 64) via v_wmma_f32_16x16x4_f32,
// then nbr_sumed = sum_m sigmoid(filt)*softplus(core),
// atom_out = softplus(atom_fea + nbr_sumed).
// W_a and the nbr_fea tile are staged into LDS by the Tensor Data Mover.
// ---------------------------------------------------------------------------
__global__ __launch_bounds__(128) void cg_atom_pass(
    const float* __restrict__ atom_fea, const float* __restrict__ nbr_fea,
    const float* __restrict__ state_fea, const float* __restrict__ W_a,
    const float* __restrict__ b_a, const int* __restrict__ atom_nbr_idx,
    const int* __restrict__ node_atom_idx, float* __restrict__ atom_out)
{
  __shared__ __align__(16) float sW[112 * 64];          // W_a, row-major [k][64]
  __shared__ __align__(16) float sNbrFea[16][MN][33];   // nbr_fea tile (TDM pad -> stride 33)
  __shared__ float sSelf[16][33];                       // atom_fea for the 16 atoms
  __shared__ float sState[16][17];                      // gathered state_fea
  __shared__ float sNbrAtom[16][MN][33];                // gathered neighbor atom features
  __shared__ int   sIdx[16][MN];
  __shared__ float sRed[4][16][32];                     // per-wave gated partial sums

  const int tid  = threadIdx.x;
  const int base = blockIdx.x * 16;

  // Async TDM staging (wave 0 issues; TDM ignores EXEC, one issue per block).
  if (tid < 32) {
    tdm_load_1d((unsigned int)(uintptr_t)&sW[0], W_a, 112 * 64, 0);
    tdm_load_1d((unsigned int)(uintptr_t)&sNbrFea[0][0][0],
                nbr_fea + (size_t)base * MN * 32, 16 * MN * 32, 1);
  }

  for (int i = tid; i < 16 * 32; i += 128) {
    int a = i >> 5, c = i & 31;
    sSelf[a][c] = atom_fea[(size_t)(base + a) * 32 + c];
  }
  for (int i = tid; i < 16 * 16; i += 128) {
    int a = i >> 4, q = i & 15;
    sState[a][q] = state_fea[(size_t)node_atom_idx[base + a] * 16 + q];
  }
  for (int i = tid; i < 16 * MN; i += 128) {
    int a = i / MN, m = i % MN;
    sIdx[a][m] = atom_nbr_idx[(size_t)(base + a) * MN + m];
  }
  __syncthreads();  // sIdx ready
  for (int i = tid; i < 16 * MN * 32; i += 128) {
    int p = i >> 5, c = i & 31;
    int a = p / MN, m = p % MN;
    sNbrAtom[a][m][c] = atom_fea[(size_t)sIdx[a][m] * 32 + c];
  }
  if (tid < 32) __builtin_amdgcn_s_wait_tensorcnt(0);  // TDM tiles landed
  __syncthreads();

  const int wave = tid >> 5;
  const int lane = tid & 31;
  const int row  = lane & 15;  // A-matrix row / B,C column-within-tile
  const int hi   = lane >> 4;  // selects K pair (A/B) and M+8 (C)

  const float bf0 = b_a[row], bf1 = b_a[16 + row];
  const float bc0 = b_a[32 + row], bc1 = b_a[48 + row];

  v8f sum0 = {0.f,0.f,0.f,0.f,0.f,0.f,0.f,0.f};
  v8f sum1 = {0.f,0.f,0.f,0.f,0.f,0.f,0.f,0.f};

  for (int mm = 0; mm < 3; ++mm) {
    const int m = wave * 3 + mm;
    v8f a0 = {0.f,0.f,0.f,0.f,0.f,0.f,0.f,0.f};
    v8f a1 = {0.f,0.f,0.f,0.f,0.f,0.f,0.f,0.f};
    v8f a2 = {0.f,0.f,0.f,0.f,0.f,0.f,0.f,0.f};
    v8f a3 = {0.f,0.f,0.f,0.f,0.f,0.f,0.f,0.f};

    for (int ks = 0; ks < 28; ++ks) {
      const int k0 = 4 * ks + 2 * hi;
      // A fragment: concat[ self(32) | gathered nbr atom(32) | nbr_fea(32) | state(16) ]
      float ax, ay;
      if (k0 < 32)      { ax = sSelf[row][k0];          ay = sSelf[row][k0 + 1]; }
      else if (k0 < 64) { ax = sNbrAtom[row][m][k0-32]; ay = sNbrAtom[row][m][k0-31]; }
      else if (k0 < 96) { ax = sNbrFea[row][m][k0-64];  ay = sNbrFea[row][m][k0-63]; }
      else              { ax = sState[row][k0-96];      ay = sState[row][k0-95]; }
      v2f A; A.x = ax; A.y = ay;

      const int wb = k0 * 64 + row;  // W_a[k0][col], next row +64
      v2f B0, B1, B2, B3;
      B0.x = sW[wb +  0]; B0.y = sW[wb + 64 +  0];
      B1.x = sW[wb + 16]; B1.y = sW[wb + 64 + 16];
      B2.x = sW[wb + 32]; B2.y = sW[wb + 64 + 32];
      B3.x = sW[wb + 48]; B3.y = sW[wb + 64 + 48];

      a0 = __builtin_amdgcn_wmma_f32_16x16x4_f32(false, A, false, B0, (short)0, a0, false, false);
      a1 = __builtin_amdgcn_wmma_f32_16x16x4_f32(false, A, false, B1, (short)0, a1, false, false);
      a2 = __builtin_amdgcn_wmma_f32_16x16x4_f32(false, A, false, B2, (short)0, a2, false, false);
      a3 = __builtin_amdgcn_wmma_f32_16x16x4_f32(false, A, false, B3, (short)0, a3, false, false);
    }
    // gate: filt = cols 0..31 (tiles 0,1), core = cols 32..63 (tiles 2,3)
    #pragma unroll
    for (int j = 0; j < 8; ++j) {
      sum0[j] += sig_(a0[j] + bf0) * sp_(a2[j] + bc0);
      sum1[j] += sig_(a1[j] + bf1) * sp_(a3[j] + bc1);
    }
  }

  // C layout: vgpr j, lanes 0-15 -> M=j, lanes 16-31 -> M=j+8; N = t*16 + row
  #pragma unroll
  for (int j = 0; j < 8; ++j) {
    sRed[wave][j + 8 * hi][row]      = sum0[j];
    sRed[wave][j + 8 * hi][16 + row] = sum1[j];
  }
  __syncthreads();
  for (int i = tid; i < 512; i += 128) {
    int a = i >> 5, c = i & 31;
    float s = sRed[0][a][c] + sRed[1][a][c] + sRed[2][a][c] + sRed[3][a][c];
    atom_out[(size_t)(base + a) * 32 + c] =
        sp_(atom_fea[(size_t)(base + a) * 32 + c] + s);
  }
}

// ---------------------------------------------------------------------------
// Kernel B: one wave per atom.  Streams nbr_fea (also writes the pass-through
// nbr_fea output), gathers atom_out[nbr], computes the rank-2 reduction of
// _embed2:  a_c = sum_m smooth_m R[m,c], b_c = sum_m R[m,c],
// alpha = sum a^2, beta = sum a*b, delta = sum b^2; atomic segment sums.
// ---------------------------------------------------------------------------
__global__ __launch_bounds__(256) void cg_edge_pass(
    const float* __restrict__ nbr_fea, const float* __restrict__ state_fea,
    const float* __restrict__ atom_out, const int* __restrict__ atom_nbr_idx,
    const int* __restrict__ node_atom_idx, float* __restrict__ out_nbr,
    float* __restrict__ ws, int Bseg)
{
  const int lane = threadIdx.x & 31;
  const int n    = (int)((blockIdx.x * blockDim.x + threadIdx.x) >> 5);
  const int seg  = node_atom_idx[n];
  const float st = (lane < 16) ? state_fea[(size_t)seg * 16 + lane] : 0.f;

  float a1 = 0.f, b1 = 0.f, a2 = 0.f, b2 = 0.f, sms = 0.f;
  for (int m = 0; m < MN; ++m) {
    size_t off = ((size_t)n * MN + m) * 32 + lane;
    float v = nbr_fea[off];
    out_nbr[off] = v;                       // output #2 (pass-through)
    float sm = __shfl(v, 0, 32);            // smooth = channel 0
    int idx = atom_nbr_idx[n * MN + m];
    float ao = atom_out[(size_t)idx * 32 + lane];
    b1 += v;  a1 = fmaf(sm, v, a1);
    b2 += ao; a2 = fmaf(sm, ao, a2);
    sms += sm;
  }
  // state channels (broadcast over m): a = sms*st, b = 12*st, lanes 0..15 only
  float a3 = (lane < 16) ? sms * st  : 0.f;
  float b3 = (lane < 16) ? 12.f * st : 0.f;
  float al = a1 * a1 + a2 * a2 + a3 * a3;
  float be = a1 * b1 + a2 * b2 + a3 * b3;
  float de = b1 * b1 + b2 * b2 + b3 * b3;
  for (int o = 16; o > 0; o >>= 1) {
    al += __shfl_xor(al, o, 32);
    be += __shfl_xor(be, o, 32);
    de += __shfl_xor(de, o, 32);
  }
  if (lane == 0) {
    atomicAdd(&ws[seg], al);
    atomicAdd(&ws[Bseg + seg], be);
    atomicAdd(&ws[2 * Bseg + seg], de);
    atomicAdd(&ws[3 * Bseg + seg], 1.0f);
  }
}

// ---------------------------------------------------------------------------
// Kernel C: per-segment tail.  q_i[j] = sum_{k,q} P_i[k*16+q] * W_s[k*16+q][j]
// with P1 = S(x)S16, P2 = S(x)b16 + b_e(x)S16, P3 = b_e(x)b16, S = colsum(W_e).
// state_out = softplus(sigmoid(pf)*pc).
// ---------------------------------------------------------------------------
__global__ __launch_bounds__(32) void cg_state_pass(
    const float* __restrict__ W_e, const float* __restrict__ b_e,
    const float* __restrict__ W_s, const float* __restrict__ b_s,
    const float* __restrict__ ws, float* __restrict__ out_state, int Bseg)
{
  __shared__ float S[32], BE[32], q1[32], q2[32], q3[32];
  const int t = threadIdx.x;
  float s = 0.f;
  for (int i = 0; i < 32; ++i) s += W_e[i * 32 + t];
  S[t] = s; BE[t] = b_e[t];
  __syncthreads();
  float c1 = 0.f, c2 = 0.f, c3 = 0.f;
  for (int k = 0; k < 32; ++k) {
    float Sk = S[k], bk = BE[k];
    for (int q = 0; q < 16; ++q) {
      float w = W_s[(size_t)((k << 4) + q) * 32 + t];
      float Sq = S[q], bq = BE[q];
      c1 = fmaf(Sk * Sq, w, c1);
      c2 = fmaf(Sk * bq + bk * Sq, w, c2);
      c3 = fmaf(bk * bq, w, c3);
    }
  }
  q1[t] = c1; q2[t] = c2; q3[t] = c3;
  __syncthreads();
  const int sgi = blockIdx.x * 32 + t;
  float cnt = fmaxf(ws[3 * Bseg + sgi], 1.0f);
  float ma = ws[sgi] / cnt;
  float mb = ws[Bseg + sgi] / cnt;
  float md = ws[2 * Bseg + sgi] / cnt;
  #pragma unroll
  for (int q = 0; q < 16; ++q) {
    float pf = ma * q1[q]      + mb * q2[q]      + md * q3[q]      + b_s[q];
    float pc = ma * q1[q + 16] + mb * q2[q + 16] + md * q3[q + 16] + b_s[q + 16];
    out_state[(size_t)sgi * 16 + q] = sp_(sig_(pf) * pc);
  }
}

extern "C" void kernel_launch(void* const* d_in, const int* in_sizes, int n_in,
                              void* d_out, int out_size, void* d_ws, size_t ws_size,
                              hipStream_t stream) {
  const float* atom_fea      = (const float*)d_in[0];
  const float* nbr_fea       = (const float*)d_in[1];
  const float* state_fea     = (const float*)d_in[2];
  const float* W_a           = (const float*)d_in[3];
  const float* b_a           = (const float*)d_in[4];
  const float* W_e           = (const float*)d_in[5];
  const float* b_e           = (const float*)d_in[6];
  const float* W_s           = (const float*)d_in[7];
  const float* b_s           = (const float*)d_in[8];
  const int*   atom_nbr_idx  = (const int*)d_in[9];
  const int*   node_atom_idx = (const int*)d_in[10];

  const int N    = in_sizes[0] / 32;   // 65536
  const int Bseg = in_sizes[2] / 16;   // 1024

  float* out_atom  = (float*)d_out;
  float* out_nbr   = out_atom + (size_t)N * 32;
  float* out_state = out_nbr + (size_t)N * MN * 32;
  float* ws        = (float*)d_ws;     // [alpha | beta | delta | count], Bseg each

  (void)hipMemsetAsync(ws, 0, (size_t)4 * Bseg * sizeof(float), stream);

  cg_atom_pass<<<N / 16, 128, 0, stream>>>(atom_fea, nbr_fea, state_fea, W_a, b_a,
                                           atom_nbr_idx, node_atom_idx, out_atom);
  cg_edge_pass<<<N / 8, 256, 0, stream>>>(nbr_fea, state_fea, out_atom,
                                          atom_nbr_idx, node_atom_idx, out_nbr,
                                          ws, Bseg);
  cg_state_pass<<<Bseg / 32, 32, 0, stream>>>(W_e, b_e, W_s, b_s, ws, out_state, Bseg);
}